// ScaledDotProductAttention_16947940950147
// MI455X (gfx1250) — compile-verified
//
#include <hip/hip_runtime.h>
#include <math.h>

#define N_FRAG  8192
#define DMODEL  256
#define DK      128
#define NUM_MOL 512

typedef __attribute__((ext_vector_type(2))) float v2f;
typedef __attribute__((ext_vector_type(8))) float v8f;

// ---------------------------------------------------------------------------
// QKV projection: Out[8192,128] = X[8192,256] @ W[256,128] + bias[128]
// One wave computes a 16x128 output strip with V_WMMA_F32_16X16X4_F32,
// accumulating over K=256 in steps of 4 (full f32 precision).
// ---------------------------------------------------------------------------
__global__ __launch_bounds__(256) void qkv_gemm_wmma(
    const float* __restrict__ X, const float* __restrict__ W,
    const float* __restrict__ bias, float* __restrict__ Out) {
  const int t     = threadIdx.x;
  const int lane  = t & 31;
  const int wave  = (blockIdx.x * 256 + t) >> 5;   // 0..511 row tiles
  const int m0    = wave * 16;
  const int mlane = lane & 15;                     // M within tile for A, N for B/C
  const int khalf = lane >> 4;                     // selects K pair {0,1} vs {2,3}

  v8f acc[8] = {};                                 // 8 tiles of 16x16 f32 = 64 VGPRs

  const float* __restrict__ xrow = X + (m0 + mlane) * DMODEL;

  for (int kk = 0; kk < DMODEL; kk += 4) {
    const int k0 = kk + khalf * 2;
    // A fragment 16x4: lane holds A[m][k0], A[m][k0+1]  (contiguous 8B load)
    v2f a;
    a.x = xrow[k0];
    a.y = xrow[k0 + 1];
#pragma unroll
    for (int n = 0; n < 8; ++n) {
      const int col = n * 16 + mlane;
      // B fragment 4x16: lane holds W[k0][col], W[k0+1][col]
      v2f b;
      b.x = W[k0 * DK + col];
      b.y = W[(k0 + 1) * DK + col];
      acc[n] = __builtin_amdgcn_wmma_f32_16x16x4_f32(
          /*neg_a=*/false, a, /*neg_b=*/false, b,
          /*c_mod=*/(short)0, acc[n], /*reuse_a=*/false, /*reuse_b=*/false);
    }
  }

  // C/D layout: VGPR r -> M = m0 + r + 8*(lane>=16), N = n*16 + (lane&15)
  const int mrow = m0 + (lane >> 4) * 8;
#pragma unroll
  for (int n = 0; n < 8; ++n) {
    const int ncol = n * 16 + mlane;
    const float bb = bias[ncol];
#pragma unroll
    for (int r = 0; r < 8; ++r) {
      Out[(mrow + r) * DK + ncol] = acc[n][r] + bb;
    }
  }
}

// ---------------------------------------------------------------------------
// Block-diagonal attention + per-molecule pooling.
// One 128-thread workgroup per molecule. molecule_index is sorted, so each
// molecule is a contiguous row range found via binary search. Online softmax
// per row; per-molecule sum accumulated in LDS (ds_add_f32), one store each.
// ---------------------------------------------------------------------------
__global__ __launch_bounds__(128) void attn_pool(
    const float* __restrict__ Q, const float* __restrict__ K,
    const float* __restrict__ V, const int* __restrict__ mol,
    float* __restrict__ out) {
  const int m    = blockIdx.x;
  const int t    = threadIdx.x;
  const int lane = t & 31;
  const int wv   = t >> 5;     // wave 0..3

  __shared__ float acc_s[DK];
  if (t < DK) acc_s[t] = 0.0f;
  __syncthreads();

  // lower_bound(mol, m) and lower_bound(mol, m+1)
  int lo = 0, hi = N_FRAG;
  while (lo < hi) { int mid = (lo + hi) >> 1; if (mol[mid] < m) lo = mid + 1; else hi = mid; }
  const int start = lo;
  hi = N_FRAG;
  while (lo < hi) { int mid = (lo + hi) >> 1; if (mol[mid] <= m) lo = mid + 1; else hi = mid; }
  const int end = lo;

  const float inv_scale = 0.08838834764831845f;  // 1/sqrt(128)

  for (int i = start + wv; i < end; i += 4) {
    const float* __restrict__ qi = Q + i * DK;
    const float q0 = qi[lane], q1 = qi[lane + 32], q2 = qi[lane + 64], q3 = qi[lane + 96];
    float mrun = -3.0e38f, lrun = 0.0f;
    float a0 = 0.f, a1 = 0.f, a2 = 0.f, a3 = 0.f;

    for (int j = start; j < end; ++j) {
      const float* __restrict__ kj = K + j * DK;
      float p = q0 * kj[lane] + q1 * kj[lane + 32] + q2 * kj[lane + 64] + q3 * kj[lane + 96];
      // wave32 all-lanes reduction
      p += __shfl_xor(p, 16, 32);
      p += __shfl_xor(p, 8, 32);
      p += __shfl_xor(p, 4, 32);
      p += __shfl_xor(p, 2, 32);
      p += __shfl_xor(p, 1, 32);
      const float s    = p * inv_scale;
      const float mnew = fmaxf(mrun, s);
      const float corr = __expf(mrun - mnew);
      const float w    = __expf(s - mnew);
      lrun = lrun * corr + w;
      const float* __restrict__ vj = V + j * DK;
      a0 = a0 * corr + w * vj[lane];
      a1 = a1 * corr + w * vj[lane + 32];
      a2 = a2 * corr + w * vj[lane + 64];
      a3 = a3 * corr + w * vj[lane + 96];
      mrun = mnew;
    }

    const float inv = 1.0f / lrun;
    atomicAdd(&acc_s[lane],      a0 * inv);
    atomicAdd(&acc_s[lane + 32], a1 * inv);
    atomicAdd(&acc_s[lane + 64], a2 * inv);
    atomicAdd(&acc_s[lane + 96], a3 * inv);
  }

  __syncthreads();
  if (t < DK) out[m * DK + t] = acc_s[t];   // zeros for empty molecules
}

// ---------------------------------------------------------------------------
extern "C" void kernel_launch(void* const* d_in, const int* in_sizes, int n_in,
                              void* d_out, int out_size, void* d_ws, size_t ws_size,
                              hipStream_t stream) {
  const float* X   = (const float*)d_in[0];
  const int*   mol = (const int*)d_in[1];
  const float* Wq  = (const float*)d_in[2];
  const float* bq  = (const float*)d_in[3];
  const float* Wk  = (const float*)d_in[4];
  const float* bk  = (const float*)d_in[5];
  const float* Wv  = (const float*)d_in[6];
  const float* bv  = (const float*)d_in[7];

  float* Q  = (float*)d_ws;                 // 8192*128 f32 = 4 MB
  float* Kp = Q + (size_t)N_FRAG * DK;      // +4 MB
  float* Vp = Kp + (size_t)N_FRAG * DK;     // +4 MB (ws needs >= 12 MB)

  qkv_gemm_wmma<<<64, 256, 0, stream>>>(X, Wq, bq, Q);
  qkv_gemm_wmma<<<64, 256, 0, stream>>>(X, Wk, bk, Kp);
  qkv_gemm_wmma<<<64, 256, 0, stream>>>(X, Wv, bv, Vp);

  attn_pool<<<NUM_MOL, 128, 0, stream>>>(Q, Kp, Vp, mol, (float*)d_out);
}